// Model_26053271617936
// MI455X (gfx1250) — compile-verified
//
#include <hip/hip_runtime.h>

// B=4, T=2048, C=256, M=2, E=16
// out[b,t,k,e] = lse[b,t] if e == label_parts[k,0] else -inf
// lse[b,t] = logsumexp_c( sum_m (score[b,t,c,m]-mean[c,m])/std[c,m]*scale[c,m] + shift[c] )

typedef float v2f __attribute__((ext_vector_type(2)));
typedef float v8f __attribute__((ext_vector_type(8)));

#define TILE_T 16
#define CCH    256
#define EPAD   260   // row stride in dwords: 260 % 64 == 4 -> t*4 distinct banks for t<16

__global__ __launch_bounds__(256) void lse_event_kernel(
    const float* __restrict__ score,   // (B*T, C, 2)
    const float* __restrict__ mean,    // (C, 2)
    const float* __restrict__ stdv,    // (C, 2)
    const float* __restrict__ scale,   // (C, 2)
    const float* __restrict__ shift,   // (C,)
    const int*   __restrict__ labels,  // (C, 2)
    float* __restrict__ out)           // (B*T, C, 16)
{
    __shared__ float e_tile[TILE_T][EPAD];   // s values, then exp values (in place)
    __shared__ float maxpart[TILE_T][16];
    __shared__ float rowmax[TILE_T];
    __shared__ float partial[TILE_T][8];     // per-wave WMMA row-sum partials
    __shared__ float lse_s[TILE_T];
    __shared__ int   tgt_l[CCH];

    const int tid  = threadIdx.x;            // 0..255
    const int bt0  = blockIdx.x * TILE_T;    // tile of 16 consecutive (b,t)

    // ---- Phase 1: per-channel params (thread tid == channel c) ----
    const int c = tid;
    const float w0   = scale[2*c + 0] / stdv[2*c + 0];
    const float w1   = scale[2*c + 1] / stdv[2*c + 1];
    const float bias = shift[c] - mean[2*c + 0] * w0 - mean[2*c + 1] * w1;
    tgt_l[c] = labels[2*c];                  // label_parts[c, 0]

    // ---- Phase 2: s[t][c] into LDS (coalesced float2 loads) ----
    const float2* sp = reinterpret_cast<const float2*>(score);
    #pragma unroll
    for (int t = 0; t < TILE_T; ++t) {
        float2 sc = sp[(size_t)(bt0 + t) * CCH + c];
        e_tile[t][c] = fmaf(sc.y, w1, fmaf(sc.x, w0, bias));
    }
    __syncthreads();

    // ---- Phase 3: row max over 256 channels (swizzled, conflict-free) ----
    const int t16 = tid >> 4;                // t handled by this thread
    const int q   = tid & 15;                // 16 threads per t, 16 channels each
    float sv[16];
    float mx = -3.0e38f;
    #pragma unroll
    for (int j = 0; j < 16; ++j) {
        sv[j] = e_tile[t16][q * 16 + ((j + q) & 15)];
        mx = fmaxf(mx, sv[j]);
    }
    maxpart[t16][q] = mx;
    __syncthreads();
    if (tid < TILE_T) {
        float m = maxpart[tid][0];
        #pragma unroll
        for (int j = 1; j < 16; ++j) m = fmaxf(m, maxpart[tid][j]);
        rowmax[tid] = m;
    }
    __syncthreads();

    // ---- Phase 4: exp(s - rowmax) written back in place ----
    const float rm = rowmax[t16];
    #pragma unroll
    for (int j = 0; j < 16; ++j)
        e_tile[t16][q * 16 + ((j + q) & 15)] = __expf(sv[j] - rm);
    __syncthreads();

    // ---- Phase 5: WMMA row-sum. Wave w reduces channels [w*32, w*32+32).
    // D = A(16x4) x ones(4x16) + C  ->  D[t,n] = C[t,n] + sum_k A[t,k]  (exact fp32)
    // A layout: lanes 0-15 supply K=0,1 (VGPR0,VGPR1); lanes 16-31 supply K=2,3.
    const int wave = tid >> 5;
    const int lane = tid & 31;
    const int m    = lane & 15;              // matrix row == timestep
    const int koff = (lane < 16) ? 0 : 2;
    const v2f ones = {1.0f, 1.0f};
    v8f acc = {0.f, 0.f, 0.f, 0.f, 0.f, 0.f, 0.f, 0.f};
    #pragma unroll
    for (int i = 0; i < 8; ++i) {
        const int kk = wave * 32 + i * 4 + koff;
        v2f a;
        a.x = e_tile[m][kk];
        a.y = e_tile[m][kk + 1];
        acc = __builtin_amdgcn_wmma_f32_16x16x4_f32(
            /*neg_a=*/false, a, /*neg_b=*/false, ones,
            /*c_mod=*/(short)0, acc, /*reuse_a=*/false, /*reuse_b=*/false);
    }
    // D column 0: lane 0 holds rows 0..7 (VGPR r -> D[r,0]); lane 16 holds rows 8..15
    if (lane == 0) {
        #pragma unroll
        for (int r = 0; r < 8; ++r) partial[r][wave] = acc[r];
    } else if (lane == 16) {
        #pragma unroll
        for (int r = 0; r < 8; ++r) partial[r + 8][wave] = acc[r];
    }
    __syncthreads();

    if (tid < TILE_T) {
        float ssum = 0.f;
        #pragma unroll
        for (int w = 0; w < 8; ++w) ssum += partial[tid][w];
        lse_s[tid] = rowmax[tid] + __logf(ssum);
    }
    __syncthreads();

    // ---- Phase 6: stream output. 16 t * 4096 floats = 16384 float4 per block.
    const float NEG_INF = __int_as_float(0xFF800000u);
    float4* op = reinterpret_cast<float4*>(out);
    #pragma unroll 4
    for (int i = 0; i < 64; ++i) {
        const int chunk = i * 256 + tid;     // 0..16383
        const int t   = chunk >> 10;         // 1024 float4 per t
        const int rem = chunk & 1023;
        const int k   = rem >> 2;            // label index
        const int e0  = (rem & 3) << 2;      // first of 4 event slots
        const float val = lse_s[t];
        const int tgt = tgt_l[k];
        float4 v;
        v.x = (tgt == e0 + 0) ? val : NEG_INF;
        v.y = (tgt == e0 + 1) ? val : NEG_INF;
        v.z = (tgt == e0 + 2) ? val : NEG_INF;
        v.w = (tgt == e0 + 3) ? val : NEG_INF;
        op[(size_t)(bt0 + t) * 1024 + rem] = v;
    }
}

extern "C" void kernel_launch(void* const* d_in, const int* in_sizes, int n_in,
                              void* d_out, int out_size, void* d_ws, size_t ws_size,
                              hipStream_t stream) {
    (void)in_sizes; (void)n_in; (void)out_size; (void)d_ws; (void)ws_size;
    const float* score  = (const float*)d_in[0];  // (4,2048,256,2)
    const float* mean   = (const float*)d_in[1];  // (256,2)
    const float* stdv   = (const float*)d_in[2];  // (256,2)
    const float* scale  = (const float*)d_in[3];  // (256,2)
    const float* shift  = (const float*)d_in[4];  // (256,)
    const int*   labels = (const int*)d_in[5];    // (256,2)
    float* out = (float*)d_out;                   // (4,2048,256,16)

    const int BT = 4 * 2048;
    dim3 grid(BT / TILE_T);                       // 512 blocks
    dim3 block(256);                              // 8 waves (wave32)
    lse_event_kernel<<<grid, block, 0, stream>>>(score, mean, stdv, scale,
                                                 shift, labels, out);
}